// Net_27238682592011
// MI455X (gfx1250) — compile-verified
//
#include <hip/hip_runtime.h>
#include <hip/hip_bf16.h>
#include <math.h>

typedef __attribute__((ext_vector_type(2))) float v2f;
typedef __attribute__((ext_vector_type(8))) float v8f;

// ---------------------------------------------------------------------------
// K0: init accumulators. deg starts at 1.0 (self-loop weight), agg1/out2 = 0.
// ---------------------------------------------------------------------------
__global__ void init_kernel(float* __restrict__ agg1, float* __restrict__ deg,
                            float* __restrict__ out2, int n) {
    int i = blockIdx.x * blockDim.x + threadIdx.x;
    int tot = n * 16;
    if (i < tot) agg1[i] = 0.0f;
    if (i < n) deg[i] = 1.0f;           // self-loop contributes weight 1
    if (i < 2 * n) out2[i] = 0.0f;
}

// ---------------------------------------------------------------------------
// K1: deg[col[e]] += w[e]   (segment_sum over targets)
// ---------------------------------------------------------------------------
__global__ void degree_kernel(const int* __restrict__ col,
                              const float* __restrict__ w,
                              float* __restrict__ deg, int ne) {
    int e = blockIdx.x * blockDim.x + threadIdx.x;
    if (e >= ne) return;
    atomicAdd(&deg[col[e]], w[e]);
}

// ---------------------------------------------------------------------------
// K2: deg -> dinv in place: dinv = deg > 0 ? rsqrt(deg) : 0
// ---------------------------------------------------------------------------
__global__ void dinv_kernel(float* __restrict__ deg, int n) {
    int i = blockIdx.x * blockDim.x + threadIdx.x;
    if (i >= n) return;
    float d = deg[i];
    deg[i] = (d > 0.0f) ? rsqrtf(d) : 0.0f;
}

// ---------------------------------------------------------------------------
// K3: hx = x @ W1  via V_WMMA_F32_16X16X4_F32.
// One wave32 per 16-row tile; K=11 padded to 12 -> 3 WMMAs.
// A layout : lane L holds row M=L%16, VGPR0=K(k0+2*(L/16)), VGPR1=K+1
// B layout : lane L holds col N=L%16, VGPR0=row k0+2*(L/16), VGPR1=row+1
// C/D      : lane L, VGPR v -> M = v + 8*(L/16), N = L%16
// ---------------------------------------------------------------------------
__global__ void gemm1_wmma(const float* __restrict__ x,
                           const float* __restrict__ W1,
                           float* __restrict__ hx, int n) {
    const int lane  = threadIdx.x & 31;
    const int wave  = blockIdx.x * (blockDim.x >> 5) + (threadIdx.x >> 5);
    const int row0  = wave * 16;
    if (row0 >= n) return;                 // wave-uniform: EXEC stays all-1s
    const int m     = lane & 15;
    const int khalf = lane >> 4;

    v8f acc = {};
    const int grow = row0 + m;             // n % 16 == 0 -> always in range
    #pragma unroll
    for (int k0 = 0; k0 < 12; k0 += 4) {
        const int ka = k0 + khalf * 2;
        v2f a, b;
        a.x = (ka     < 11) ? x[grow * 11 + ka]     : 0.0f;
        a.y = (ka + 1 < 11) ? x[grow * 11 + ka + 1] : 0.0f;
        b.x = (ka     < 11) ? W1[ka * 16 + m]       : 0.0f;
        b.y = (ka + 1 < 11) ? W1[(ka + 1) * 16 + m] : 0.0f;
        acc = __builtin_amdgcn_wmma_f32_16x16x4_f32(
            /*neg_a=*/false, a, /*neg_b=*/false, b,
            /*c_mod=*/(short)0, acc, /*reuse_a=*/false, /*reuse_b=*/false);
    }
    #pragma unroll
    for (int v = 0; v < 8; ++v)
        hx[(size_t)(row0 + v + 8 * khalf) * 16 + m] = acc[v];
}

// ---------------------------------------------------------------------------
// K4: layer-1 edge aggregation.  agg1[col] += dinv[r]*w*dinv[c] * hx[row]
// hx (32MB) and dinv (2MB) are L2-resident on MI455X (192MB L2).
// ---------------------------------------------------------------------------
__global__ void agg1_kernel(const int* __restrict__ row,
                            const int* __restrict__ col,
                            const float* __restrict__ w,
                            const float* __restrict__ dinv,
                            const float* __restrict__ hx,
                            float* __restrict__ agg1, int ne) {
    int e = blockIdx.x * blockDim.x + threadIdx.x;
    if (e >= ne) return;
    const int r = row[e];
    const int c = col[e];
    const float nrm = dinv[r] * w[e] * dinv[c];
    const float4* hp = (const float4*)(hx + (size_t)r * 16);
    float* op = agg1 + (size_t)c * 16;
    #pragma unroll
    for (int q = 0; q < 4; ++q) {
        float4 hv = hp[q];
        atomicAdd(op + q * 4 + 0, nrm * hv.x);
        atomicAdd(op + q * 4 + 1, nrm * hv.y);
        atomicAdd(op + q * 4 + 2, nrm * hv.z);
        atomicAdd(op + q * 4 + 3, nrm * hv.w);
    }
}

// ---------------------------------------------------------------------------
// K5: per-node: h1 = relu(agg1 + dinv^2*hx + b1); h2 = h1 @ W2 (16x2, fused)
// ---------------------------------------------------------------------------
__global__ void node1_kernel(const float* __restrict__ agg1,
                             const float* __restrict__ hx,
                             const float* __restrict__ dinv,
                             const float* __restrict__ b1,
                             const float* __restrict__ W2,
                             float* __restrict__ h2, int n) {
    int i = blockIdx.x * blockDim.x + threadIdx.x;
    if (i >= n) return;
    const float di = dinv[i];
    const float self = di * di;
    float s0 = 0.0f, s1 = 0.0f;
    #pragma unroll
    for (int f = 0; f < 16; ++f) {
        float v = agg1[(size_t)i * 16 + f] + self * hx[(size_t)i * 16 + f] + b1[f];
        v = v > 0.0f ? v : 0.0f;
        s0 = fmaf(v, W2[f * 2 + 0], s0);
        s1 = fmaf(v, W2[f * 2 + 1], s1);
    }
    h2[(size_t)i * 2 + 0] = s0;
    h2[(size_t)i * 2 + 1] = s1;
}

// ---------------------------------------------------------------------------
// K6: layer-2 edge aggregation, accumulating directly into d_out (N x 2)
// ---------------------------------------------------------------------------
__global__ void agg2_kernel(const int* __restrict__ row,
                            const int* __restrict__ col,
                            const float* __restrict__ w,
                            const float* __restrict__ dinv,
                            const float* __restrict__ h2,
                            float* __restrict__ out2, int ne) {
    int e = blockIdx.x * blockDim.x + threadIdx.x;
    if (e >= ne) return;
    const int r = row[e];
    const int c = col[e];
    const float nrm = dinv[r] * w[e] * dinv[c];
    const float2 hv = ((const float2*)h2)[r];
    atomicAdd(&out2[(size_t)c * 2 + 0], nrm * hv.x);
    atomicAdd(&out2[(size_t)c * 2 + 1], nrm * hv.y);
}

// ---------------------------------------------------------------------------
// K7: add self term + b2, then 2-class log_softmax in place.
// ---------------------------------------------------------------------------
__global__ void out_kernel(float* __restrict__ out2,
                           const float* __restrict__ h2,
                           const float* __restrict__ dinv,
                           const float* __restrict__ b2, int n) {
    int i = blockIdx.x * blockDim.x + threadIdx.x;
    if (i >= n) return;
    const float di = dinv[i];
    const float self = di * di;
    float v0 = out2[(size_t)i * 2 + 0] + self * h2[(size_t)i * 2 + 0] + b2[0];
    float v1 = out2[(size_t)i * 2 + 1] + self * h2[(size_t)i * 2 + 1] + b2[1];
    const float m = fmaxf(v0, v1);
    const float lse = m + logf(expf(v0 - m) + expf(v1 - m));
    out2[(size_t)i * 2 + 0] = v0 - lse;
    out2[(size_t)i * 2 + 1] = v1 - lse;
}

// ---------------------------------------------------------------------------
extern "C" void kernel_launch(void* const* d_in, const int* in_sizes, int n_in,
                              void* d_out, int out_size, void* d_ws, size_t ws_size,
                              hipStream_t stream) {
    const float* x   = (const float*)d_in[0];   // N x 11
    const int*   ei  = (const int*)  d_in[1];   // 2 x E
    const float* ew  = (const float*)d_in[2];   // E
    const float* W1  = (const float*)d_in[3];   // 11 x 16
    const float* b1  = (const float*)d_in[4];   // 16
    const float* W2  = (const float*)d_in[5];   // 16 x 2
    const float* b2  = (const float*)d_in[6];   // 2

    const int n = in_sizes[0] / 11;             // 500000
    const int e = in_sizes[2];                  // 16000000
    const int* row = ei;                        // edge_index[0] = source
    const int* col = ei + e;                    // edge_index[1] = target

    // Workspace layout (floats): [dinv: n][hx: 16n][agg1: 16n][h2: 2n] = 35n
    float* ws   = (float*)d_ws;
    float* dinv = ws;                           // deg, then rsqrt in place
    float* hx   = ws + (size_t)n;
    float* agg1 = ws + (size_t)n * 17;
    float* h2   = ws + (size_t)n * 33;
    float* out2 = (float*)d_out;                // agg2 accumulates into d_out

    const int B = 256;
    const int gN16  = (n * 16 + B - 1) / B;
    const int gN    = (n + B - 1) / B;
    const int gE    = (e + B - 1) / B;
    const int waves = (n + 15) / 16;            // one wave per 16-row tile
    const int gGemm = (waves + (B / 32) - 1) / (B / 32);

    init_kernel  <<<gN16, B, 0, stream>>>(agg1, dinv, out2, n);
    degree_kernel<<<gE,   B, 0, stream>>>(col, ew, dinv, e);
    dinv_kernel  <<<gN,   B, 0, stream>>>(dinv, n);
    gemm1_wmma   <<<gGemm,B, 0, stream>>>(x, W1, hx, n);
    agg1_kernel  <<<gE,   B, 0, stream>>>(row, col, ew, dinv, hx, agg1, e);
    node1_kernel <<<gN,   B, 0, stream>>>(agg1, hx, dinv, b1, W2, h2, n);
    agg2_kernel  <<<gE,   B, 0, stream>>>(row, col, ew, dinv, h2, out2, e);
    out_kernel   <<<gN,   B, 0, stream>>>(out2, h2, dinv, b2, n);
}